// DiffSortNet_65154653880715
// MI455X (gfx1250) — compile-verified
//
#include <hip/hip_runtime.h>

// ---------------------------------------------------------------------------
// DiffSort (bitonic soft sort), N=1024, BATCH=8, 55 layers.
// Outputs: x (8x1024) then X (8x1024x1024), concatenated flat in d_out.
//
// Plan (MI455X / gfx1250):
//  K1: per-batch x pass in LDS, stores alpha[layer][batch][pair] to d_ws.
//  KG: composes the last 4 layers (strides 8,4,2,1 -> within 16-col groups)
//      into dense 16x16 blocks G[batch][group] (block-diagonal product).
//  K2: per (batch, 16-row tile): identity tile in LDS, 51 layers of LDS
//      column mixing with TDM-double-buffered alpha/idx staging, then the
//      composed G applied via V_WMMA_F32_16X16X4_F32, then one
//      tensor_store_from_lds writes the 64KB tile to HBM.
// ---------------------------------------------------------------------------

typedef unsigned int uint32x4 __attribute__((ext_vector_type(4)));
typedef int          int32x4  __attribute__((ext_vector_type(4)));
typedef int          int32x8  __attribute__((ext_vector_type(8)));
typedef float        v2f      __attribute__((ext_vector_type(2)));
typedef float        v8f      __attribute__((ext_vector_type(8)));

#define NLAYERS 55
#define NB      8
#define NN      1024
#define NPAIR   512

// ---------------- TDM descriptor (CDNA5 ISA)
__device__ __forceinline__ void tdm_build(uint32x4& g0, int32x8& g1,
                                          unsigned lds_addr,
                                          unsigned long long gaddr,
                                          unsigned tensor_d0,
                                          unsigned tile_d0)
{
    // group0: count=1 | lds_addr[63:32] | global_addr[120:64] | type=2[127:126]
    g0.x = 1u;
    g0.y = lds_addr;
    g0.z = (unsigned)(gaddr & 0xffffffffull);
    g0.w = (unsigned)((gaddr >> 32) & 0x01ffffffull) | (2u << 30);
    // group1: data_size=2 (4B) @ [17:16]; tensor_dim0 @ [79:48];
    //         tile_dim0 @ [127:112]; tensor_dim0_stride @ [207:160]
    unsigned long long s0 = tensor_d0;
    g1[0] = (int)(2u << 16);
    g1[1] = (int)((tensor_d0 & 0xffffu) << 16);
    g1[2] = (int)(tensor_d0 >> 16);               // tensor_dim1 = 0
    g1[3] = (int)(tile_d0 << 16);                 // tile_dim0
    g1[4] = 0;                                    // tile_dim1 = tile_dim2 = 0 (1D)
    g1[5] = (int)(unsigned)(s0 & 0xffffffffull);
    g1[6] = (int)(unsigned)((s0 >> 32) & 0xffffull);
    g1[7] = 0;
}

__device__ __forceinline__ void tdm_load_1d(void* lds, const void* gsrc, unsigned n)
{
    uint32x4 g0; int32x8 g1;
    tdm_build(g0, g1, (unsigned)(unsigned long long)(uintptr_t)lds,
              (unsigned long long)(uintptr_t)gsrc, n, n);
    int32x4 z4 = {0, 0, 0, 0};
#if defined(__clang_major__) && (__clang_major__ >= 23)
    int32x8 z8 = {0, 0, 0, 0, 0, 0, 0, 0};
    __builtin_amdgcn_tensor_load_to_lds(g0, g1, z4, z4, z8, 0);
#else
    __builtin_amdgcn_tensor_load_to_lds(g0, g1, z4, z4, 0);
#endif
}

__device__ __forceinline__ void tdm_store_1d(void* gdst, const void* lds, unsigned n)
{
    uint32x4 g0; int32x8 g1;
    tdm_build(g0, g1, (unsigned)(unsigned long long)(uintptr_t)lds,
              (unsigned long long)(uintptr_t)gdst, n, n);
    int32x4 z4 = {0, 0, 0, 0};
#if defined(__clang_major__) && (__clang_major__ >= 23)
    int32x8 z8 = {0, 0, 0, 0, 0, 0, 0, 0};
    __builtin_amdgcn_tensor_store_from_lds(g0, g1, z4, z4, z8, 0);
#else
    __builtin_amdgcn_tensor_store_from_lds(g0, g1, z4, z4, 0);
#endif
}

// ---------------- soft compare-exchange alpha (matches reference) ------------
__device__ __forceinline__ float soft_alpha(float d)
{
    // sigmoid(d * 10 / (|d| + 1e-10)^0.25)
    float t = __powf(fabsf(d) + 1e-10f, 0.25f);
    float z = d * 10.0f / t;
    return 1.0f / (1.0f + __expf(-z));
}

// ---------------- K1: x pass + alpha dump -----------------------------------
__global__ void __launch_bounds__(512)
k1_sort_x(const float* __restrict__ vec, const int* __restrict__ idx_a,
          const int* __restrict__ idx_b, float* __restrict__ x_out,
          float* __restrict__ alpha_ws)
{
    __shared__ float xs[NN];
    const int b = blockIdx.x;
    const int t = threadIdx.x;                    // 512 threads, pair t
    xs[t]       = vec[b * NN + t];
    xs[t + 512] = vec[b * NN + t + 512];
    __syncthreads();
    for (int l = 0; l < NLAYERS; ++l) {
        const int ia = idx_a[l * NPAIR + t];
        const int ib = idx_b[l * NPAIR + t];
        float a  = xs[ia];
        float bb = xs[ib];
        float al = soft_alpha(bb - a);
        xs[ia] = bb + al * (a - bb);              // alpha*a + (1-alpha)*b
        xs[ib] = a + al * (bb - a);               // (1-alpha)*a + alpha*b
        alpha_ws[(l * NB + b) * NPAIR + t] = al;
        __syncthreads();
    }
    x_out[b * NN + t]       = xs[t];
    x_out[b * NN + t + 512] = xs[t + 512];
}

// ---------------- KG: compose layers 51..54 into 16x16 blocks G --------------
// Block 9 (layers 45..54) has no direction flips (col/1024 parity == 0), so
// layers with m=8,4,2,1 pair columns (a, a+m) inside aligned 16-col groups.
__global__ void __launch_bounds__(128)
kg_compose(const float* __restrict__ alpha_ws, float* __restrict__ g_ws)
{
    __shared__ float Gt[256];
    const int g = blockIdx.x;                     // 64 column groups
    const int b = blockIdx.y;                     // 8 batches
    const int t = threadIdx.x;                    // 128 threads
    Gt[t]       = ((t >> 4) == (t & 15)) ? 1.0f : 0.0f;
    Gt[t + 128] = (((t + 128) >> 4) == ((t + 128) & 15)) ? 1.0f : 0.0f;
    __syncthreads();
    const int q = t >> 4;                         // pair within group: 0..7
    const int r = t & 15;                         // row of G: 0..15
    for (int k = 0; k < 4; ++k) {                 // layers 51..54 (m = 8,4,2,1)
        const int lm = 3 - k;
        const int m  = 1 << lm;
        const int a_loc  = ((q >> lm) << (lm + 1)) | (q & (m - 1));
        const int b_loc  = a_loc + m;
        const int a_glob = g * 16 + a_loc;
        const int p = ((a_glob >> (lm + 1)) << lm) | (a_glob & (m - 1));
        const float al = alpha_ws[((51 + k) * NB + b) * NPAIR + p];
        float A = Gt[r * 16 + a_loc];
        float B = Gt[r * 16 + b_loc];
        Gt[r * 16 + a_loc] = B + al * (A - B);
        Gt[r * 16 + b_loc] = A + al * (B - A);
        __syncthreads();
    }
    g_ws[(b * 64 + g) * 256 + t]       = Gt[t];
    g_ws[(b * 64 + g) * 256 + t + 128] = Gt[t + 128];
}

// ---------------- K2: X tiles in LDS + TDM staging + WMMA epilogue -----------
__global__ void __launch_bounds__(512)
k2_perm(const int* __restrict__ idx_a, const int* __restrict__ idx_b,
        const float* __restrict__ alpha_ws, const float* __restrict__ g_ws,
        float* __restrict__ outX)
{
    extern __shared__ float smem[];
    float* tile  = smem;                          // 16 x 1024 = 64 KB
    float* abuf  = smem + 16 * NN;                // 2 x 512 alpha (ping-pong)
    int*   iabuf = (int*)(abuf + 2 * NPAIR);      // 2 x 512 idx_a
    int*   ibbuf = iabuf + 2 * NPAIR;             // 2 x 512 idx_b

    const int t    = threadIdx.x;                 // 512 threads = 16 waves
    const int b    = blockIdx.y;
    const int row0 = blockIdx.x * 16;

    // Kick off TDM DMA of layer-0 staging while we build the identity tile.
    if (t == 0) {
        tdm_load_1d(abuf,  alpha_ws + (0 * NB + b) * NPAIR, NPAIR);
        tdm_load_1d(iabuf, idx_a, NPAIR);
        tdm_load_1d(ibbuf, idx_b, NPAIR);
    }
    for (int i = t; i < 16 * NN; i += 512)
        tile[i] = ((row0 + (i >> 10)) == (i & (NN - 1))) ? 1.0f : 0.0f;
    if (t == 0) __builtin_amdgcn_s_wait_tensorcnt(0);
    __syncthreads();

    // Layers 0..50: LDS column mixing; TDM prefetch of next layer's staging.
    for (int l = 0; l < 51; ++l) {
        const int cur = l & 1;
        const int nxt = cur ^ 1;
        if (t == 0 && l < 50) {
            tdm_load_1d(abuf  + nxt * NPAIR, alpha_ws + ((l + 1) * NB + b) * NPAIR, NPAIR);
            tdm_load_1d(iabuf + nxt * NPAIR, idx_a + (l + 1) * NPAIR, NPAIR);
            tdm_load_1d(ibbuf + nxt * NPAIR, idx_b + (l + 1) * NPAIR, NPAIR);
        }
        const int   ia = iabuf[cur * NPAIR + t];
        const int   ib = ibbuf[cur * NPAIR + t];
        const float al = abuf[cur * NPAIR + t];
#pragma unroll
        for (int r = 0; r < 16; ++r) {
            float A = tile[r * NN + ia];
            float B = tile[r * NN + ib];
            tile[r * NN + ia] = B + al * (A - B);
            tile[r * NN + ib] = A + al * (B - A);
        }
        if (t == 0 && l < 50) __builtin_amdgcn_s_wait_tensorcnt(0);
        __syncthreads();
    }

    // Layers 51..54 as one block-diagonal GEMM: tile <- tile * G (per group).
    // V_WMMA_F32_16X16X4_F32 fragment layouts per ISA 7.12.2:
    //   A 16x4:  lane L -> M = L&15, K = 4s + 2*(L>>4) + {0,1}
    //   B 4x16:  lane L -> N = L&15, K = 4s + 2*(L>>4) + {0,1}
    //   C/D:     lane L, VGPR v -> M = v + 8*(L>>4), N = L&15
    {
        const int lane = t & 31;
        const int wave = t >> 5;
        const int mrow = lane & 15;
        const int half = lane >> 4;
        const float* Gb = g_ws + (b * 64) * 256;
#pragma unroll
        for (int k = 0; k < 4; ++k) {
            const int g = wave * 4 + k;           // 16 waves x 4 = 64 groups
            const float* G = Gb + g * 256;
            v8f acc = {};
#pragma unroll
            for (int s = 0; s < 4; ++s) {
                const int kk = 4 * s + 2 * half;
                v2f av = { tile[mrow * NN + g * 16 + kk],
                           tile[mrow * NN + g * 16 + kk + 1] };
                v2f bv = { G[kk * 16 + mrow],
                           G[(kk + 1) * 16 + mrow] };
                acc = __builtin_amdgcn_wmma_f32_16x16x4_f32(
                          false, av, false, bv, (short)0, acc, false, false);
            }
#pragma unroll
            for (int v = 0; v < 8; ++v)
                tile[(v + 8 * half) * NN + g * 16 + mrow] = acc[v];
        }
    }
    __syncthreads();

    // One 64KB TDM store of the finished tile (rows are contiguous in X).
    if (t == 0) {
        tdm_store_1d(outX + (size_t)b * NN * NN + (size_t)row0 * NN, tile, 16 * NN);
        __builtin_amdgcn_s_wait_tensorcnt(0);
    }
}

// ---------------------------------------------------------------------------
extern "C" void kernel_launch(void* const* d_in, const int* in_sizes, int n_in,
                              void* d_out, int out_size, void* d_ws, size_t ws_size,
                              hipStream_t stream)
{
    (void)in_sizes; (void)n_in; (void)out_size; (void)ws_size;
    const float* vec = (const float*)d_in[0];
    const int*   ia  = (const int*)d_in[1];     // (55, 512)
    const int*   ib  = (const int*)d_in[2];     // (55, 512)

    float* xout = (float*)d_out;                // (8, 1024)
    float* Xout = xout + NB * NN;               // (8, 1024, 1024)

    float* alpha_ws = (float*)d_ws;             // 55*8*512 floats (~900 KB)
    float* g_ws     = alpha_ws + NLAYERS * NB * NPAIR; // 8*64*256 floats (~512 KB)

    k1_sort_x<<<NB, 512, 0, stream>>>(vec, ia, ib, xout, alpha_ws);
    kg_compose<<<dim3(64, NB), 128, 0, stream>>>(alpha_ws, g_ws);

    const size_t smem = (size_t)(16 * NN + 2 * NPAIR + 4 * NPAIR) * sizeof(float);
    k2_perm<<<dim3(64, NB), 512, smem, stream>>>(ia, ib, alpha_ws, g_ws, Xout);
}